// STBIFNeuron_76742475645455
// MI455X (gfx1250) — compile-verified
//
#include <hip/hip_runtime.h>

// ST-BIF neuron, T=16 sequential scan, elementwise over [B,C,H,W]=[16,128,32,32].
// Pure HBM-bandwidth-bound streaming kernel: 256 MiB traffic -> ~11.5 us floor
// at 23.3 TB/s. float4 per thread, full T-unroll, non-temporal b128 vmem.

typedef __attribute__((ext_vector_type(4))) float v4f;

namespace {
constexpr int T = 16;
constexpr int B = 16;
constexpr int C = 128;
constexpr int H = 32;
constexpr int W = 32;
constexpr int HW = H * W;                 // 1024
constexpr int N  = B * C * H * W;         // 2,097,152 elements per timestep
constexpr int N4 = N / 4;                 // 524,288 float4 per timestep

constexpr float TWO_N      = 8.0f;        // threshold 2^3
constexpr float Q_INIT     = 4.0f;        // 2^(3-1)
constexpr float POS_MAX    = 7.0f;
constexpr float NEG_MIN    = -8.0f;
constexpr float BIAS_SCALE = TWO_N / 5.0f; // bias * 2^N / BIAS_STEPS
constexpr int   BIAS_STEPS = 5;
} // namespace

__global__ __launch_bounds__(256) void stbif_scan_kernel(
    const float* __restrict__ x,     // [T, N]
    const float* __restrict__ bias,  // [C]
    float* __restrict__ out)         // [T, N]
{
    const int tid = blockIdx.x * blockDim.x + threadIdx.x;  // float4 index
    if (tid >= N4) return;

    // channel of this float4 (HW=1024 is a multiple of 4, so all 4 elements
    // of the vector share one channel)
    const int elem = tid * 4;
    const int c    = (elem / HW) % C;
    const float bstep = bias[c] * BIAS_SCALE;

    v4f q   = {Q_INIT, Q_INIT, Q_INIT, Q_INIT};
    v4f acc = {0.0f, 0.0f, 0.0f, 0.0f};

    const v4f* __restrict__ xv = (const v4f*)x;
    v4f* __restrict__       ov = (v4f*)out;

#pragma unroll
    for (int t = 0; t < T; ++t) {
        // independent of the scan state -> compiler can hoist these loads
        v4f xt = __builtin_nontemporal_load(&xv[(size_t)t * N4 + tid]);

        if (t < BIAS_STEPS) {
            q.x += bstep; q.y += bstep; q.z += bstep; q.w += bstep;
        }
        q += xt;

        v4f cur;
#pragma unroll
        for (int i = 0; i < 4; ++i) {
            float qi = q[i];
            float ai = acc[i];
            // acc is always exactly integral (starts 0, +/-1 steps): round == id.
            // spike (q>=8) and neg (q<0) are mutually exclusive.
            float s  = (qi >= TWO_N && ai < POS_MAX) ? 1.0f : 0.0f;
            float n  = (qi < 0.0f   && ai > NEG_MIN) ? 1.0f : 0.0f;
            float cu = s - n;
            ai += cu;
            qi -= TWO_N * cu;   // q - 8*spike + 8*neg
            q[i]   = qi;
            acc[i] = ai;
            cur[i] = cu;
        }

        __builtin_nontemporal_store(cur, &ov[(size_t)t * N4 + tid]);
    }
}

extern "C" void kernel_launch(void* const* d_in, const int* in_sizes, int n_in,
                              void* d_out, int out_size, void* d_ws, size_t ws_size,
                              hipStream_t stream) {
    (void)in_sizes; (void)n_in; (void)out_size; (void)d_ws; (void)ws_size;

    const float* x    = (const float*)d_in[0];  // [T,B,C,H,W] float32
    const float* bias = (const float*)d_in[1];  // [C] float32
    float* out        = (float*)d_out;          // [T,B,C,H,W] float32

    constexpr int kBlock = 256;                 // 8 wave32 waves per block
    constexpr int kGrid  = (N4 + kBlock - 1) / kBlock;  // 2048 blocks

    stbif_scan_kernel<<<kGrid, kBlock, 0, stream>>>(x, bias, out);
}